// DepthIoULoss_82927228551956
// MI455X (gfx1250) — compile-verified
//
#include <hip/hip_runtime.h>

typedef __attribute__((ext_vector_type(2))) float v2f;
typedef __attribute__((ext_vector_type(4))) float v4f;
typedef __attribute__((ext_vector_type(8))) float v8f;

#define S_LEN   2048
#define TPB     256
#define VEC_PER_T 2          // 2 x float4 = 8 elements per thread; 256*8 = 2048

// Wave32 sum reduction through the matrix pipe:
// A(16x4): lane L, VGPR0 = K0 (L<16) / K2 (L>=16); VGPR1 zeroed.
// B(4x16) = all ones, C = 0  =>  D[m][n] = val[m] + val[m+16].
// C/D layout: lane L holds rows {v + 8*(L>=16)} in c[v], so sum(c[0..7]) gives
// half-sums; one xor-16 shuffle completes the 32-lane total (in every lane).
__device__ __forceinline__ float wave_sum32(float v) {
    v2f a; a[0] = v;    a[1] = 0.0f;
    v2f b; b[0] = 1.0f; b[1] = 1.0f;
    v8f c = {};
    c = __builtin_amdgcn_wmma_f32_16x16x4_f32(false, a, false, b,
                                              (short)0, c, false, false);
    float s = c[0] + c[1] + c[2] + c[3] + c[4] + c[5] + c[6] + c[7];
    s += __shfl_xor(s, 16, 32);
    return s;
}

__global__ __launch_bounds__(TPB) void depth_iou_rows(
        const float* __restrict__ preds,
        const float* __restrict__ targs,
        float* __restrict__ row_out) {
    __shared__ int   s_stop;
    __shared__ float s_pw[TPB * VEC_PER_T];   // .w of each float4 chunk
    __shared__ float s_tw[TPB * VEC_PER_T];
    __shared__ float s_wsum[TPB / 32];

    const int    t    = threadIdx.x;
    const size_t base = (size_t)blockIdx.x * S_LEN;
    const float* pr = preds + base;
    const float* tr = targs + base;

    if (t == 0) s_stop = 1 << 30;

    v4f p[VEC_PER_T], g[VEC_PER_T];
    int localStop = 1 << 30;
    #pragma unroll
    for (int i = 0; i < VEC_PER_T; ++i) {
        const int idx = i * TPB + t;                 // float4 chunk id 0..511
        p[i] = *(const v4f*)(pr + 4 * idx);
        g[i] = *(const v4f*)(tr + 4 * idx);
        s_pw[idx] = p[i][3];
        s_tw[idx] = g[i][3];
        #pragma unroll
        for (int c = 0; c < 4; ++c)
            if (g[i][c] == 1.0f) localStop = min(localStop, 4 * idx + c);
    }
    __syncthreads();
    if (localStop < (1 << 30)) atomicMin(&s_stop, localStop);
    __syncthreads();

    const int  stop     = s_stop;
    const bool has_stop = stop < S_LEN;

    float acc = 0.0f;
    #pragma unroll
    for (int i = 0; i < VEC_PER_T; ++i) {
        const int   idx   = i * TPB + t;
        const float prevP = (idx == 0) ? 0.0f : s_pw[idx - 1];
        const float prevT = (idx == 0) ? 0.0f : s_tw[idx - 1];
        #pragma unroll
        for (int c = 0; c < 4; ++c) {
            const int   j   = 4 * idx + c;
            const float pl  = (c == 0) ? prevP : p[i][c - 1];
            const float prt = p[i][c];
            const float tl  = (c == 0) ? prevT : g[i][c - 1];
            const float trt = g[i][c];

            const bool disjoint = (prt <= tl) || (trt <= pl);
            // full 4-sort (matches jnp.sort even for degenerate intervals)
            const float a1 = fminf(pl, prt),  a2 = fmaxf(pl, prt);
            const float b1 = fminf(tl, trt),  b2 = fmaxf(tl, trt);
            const float sd0 = fminf(a1, b1);
            const float sd3 = fmaxf(a2, b2);
            const float m1  = fmaxf(a1, b1);
            const float m2  = fminf(a2, b2);
            const float sd1 = fminf(m1, m2);
            const float sd2 = fmaxf(m1, m2);

            const float inter = disjoint ? 0.0f : (sd2 - sd1);
            const float uni   = sd3 - sd0;
            float iou = (uni > 0.0f) ? inter * __builtin_amdgcn_rcpf(uni) : 0.0f;

            const bool valid = has_stop && (j <= stop);
            acc += valid ? iou : 0.0f;
        }
    }

    const float wsum = wave_sum32(acc);
    if ((t & 31) == 0) s_wsum[t >> 5] = wsum;
    __syncthreads();
    if (t == 0) {
        float tot = 0.0f;
        #pragma unroll
        for (int w = 0; w < TPB / 32; ++w) tot += s_wsum[w];
        row_out[blockIdx.x] = has_stop ? (tot / (float)(stop + 1)) : 0.0f;
    }
}

__global__ __launch_bounds__(TPB) void final_reduce(
        const float* __restrict__ row_vals, float* __restrict__ out, int n) {
    __shared__ float s_wsum[TPB / 32];
    const int t = threadIdx.x;
    float acc = 0.0f;
    for (int idx4 = t; 4 * idx4 < n; idx4 += TPB) {
        v4f v = *(const v4f*)(row_vals + 4 * idx4);
        acc += v[0] + v[1] + v[2] + v[3];
    }
    const float wsum = wave_sum32(acc);
    if ((t & 31) == 0) s_wsum[t >> 5] = wsum;
    __syncthreads();
    if (t == 0) {
        float tot = 0.0f;
        #pragma unroll
        for (int w = 0; w < TPB / 32; ++w) tot += s_wsum[w];
        out[0] = 1.0f - tot / (float)n;
    }
}

extern "C" void kernel_launch(void* const* d_in, const int* in_sizes, int n_in,
                              void* d_out, int out_size, void* d_ws, size_t ws_size,
                              hipStream_t stream) {
    const float* preds = (const float*)d_in[0];
    const float* targs = (const float*)d_in[1];
    const int rows = in_sizes[0] / S_LEN;          // 8192
    float* row_ws = (float*)d_ws;                  // rows * 4 bytes scratch

    depth_iou_rows<<<rows, TPB, 0, stream>>>(preds, targs, row_ws);
    final_reduce<<<1, TPB, 0, stream>>>(row_ws, (float*)d_out, rows);
}